// NNFiber_38431367365148
// MI455X (gfx1250) — compile-verified
//
#include <hip/hip_runtime.h>
#include <cstdint>
#include <math.h>

// ---------------------------------------------------------------------------
// SSFM fiber sim on MI455X (gfx1250):
//   - N = 262144 = 64^3 FFT as three radix-64 stages; each stage is a complex
//     64x64 (DFT matrix) x 64xL (data) GEMM built from V_WMMA_F32_16X16X4_F32.
//   - Panel staging: TDM tensor_load_to_lds for stages 1-2 (2D tile / linear),
//     global_load_async_to_lds_b32 for the stage-3 transpose gather.
//   - Everything (55-67 MB) is L2-resident on the 192 MB L2.
// ---------------------------------------------------------------------------

#define NFFT   262144      // 2^18
#define RADIX  64
#define NLINES 4096        // NFFT / RADIX
#define PANEL  64          // lines (matmul columns) per block
#define NSIG   16          // 8 batch x 2 pol
#define BATCH  8
#define NSTEP  10
#define LOUT   32768       // (NFFT-8)/8 + 1
#define PI_F   3.14159265358979323846f

#ifndef USE_ASYNC
#define USE_ASYNC 1
#endif

#if defined(__has_builtin)
#if __has_builtin(__builtin_amdgcn_tensor_load_to_lds) && \
    __has_builtin(__builtin_amdgcn_s_wait_tensorcnt)
#define HAVE_TDM 1
#endif
#endif
#ifndef HAVE_TDM
#define HAVE_TDM 0
#endif

typedef float v2f __attribute__((ext_vector_type(2)));
typedef float v8f __attribute__((ext_vector_type(8)));
typedef unsigned int v4u __attribute__((ext_vector_type(4)));
typedef int v4i __attribute__((ext_vector_type(4)));
typedef int v8i __attribute__((ext_vector_type(8)));

__device__ __forceinline__ uint32_t lds_off(const void* p) {
  // low 32 bits of a generic pointer into LDS = byte offset within LDS window
  return (uint32_t)(uintptr_t)p;
}

#if HAVE_TDM
// Issue one TDM 2D tile load: tile1 rows of tile0 contiguous f32 elements,
// row pitch stride0 (elements), written contiguously to LDS at ldsOff.
// clang-23 builtin arity is 6: (g0, g1, g2, g3, pad(int32x8), cpol).
__device__ __forceinline__ void tdm_load_2d(const float* gptr, uint32_t ldsOff,
                                            uint32_t tile0, uint32_t tile1,
                                            uint32_t stride0)
{
  const unsigned long long ga = (unsigned long long)(uintptr_t)gptr;
  const uint32_t td0 = 1u << 20, td1 = 1u << 20;  // huge tensor dims: no OOB clamp
  v4u g0;
  g0.x = 1u;                                           // count=1 (valid), user mode
  g0.y = ldsOff;                                       // LDS byte address
  g0.z = (uint32_t)ga;                                 // global_addr[31:0]
  g0.w = (uint32_t)((ga >> 32) & 0x01FFFFFFull)        // global_addr[56:32]
         | (2u << 30);                                 // type = 2 ("image")
  v8i g1;
  g1[0] = (int)(2u << 16);                             // data_size = 4 bytes
  g1[1] = (int)((td0 & 0xFFFFu) << 16);                // tensor_dim0 low16 @ bit48
  g1[2] = (int)((td0 >> 16) | ((td1 & 0xFFFFu) << 16));// td0 hi16 | td1 lo16
  g1[3] = (int)((td1 >> 16) | (tile0 << 16));          // td1 hi16 | tile_dim0
  g1[4] = (int)tile1;                                  // tile_dim1, tile_dim2=0
  g1[5] = (int)stride0;                                // tensor_dim0_stride[31:0]
  g1[6] = 0;                                           // stride0 hi16 | stride1 lo16
  g1[7] = 0;                                           // stride1 hi32
  v4i g2; g2[0] = 1; g2[1] = 1; g2[2] = 0; g2[3] = 0;  // tensor_dim2=1, tensor_dim3=1
  v4i g3; g3[0] = 0; g3[1] = (int)(1u << 16); g3[2] = 0; g3[3] = 0; // tensor_dim4=1
  const v8i zpad = {};                                 // unused trailing group
  __builtin_amdgcn_tensor_load_to_lds(g0, g1, g2, g3, zpad, 0);
}
#endif

// ---------------------------------------------------------------------------
// One radix-64 stage of the 64^3 Cooley-Tukey factorization.
//   stage 1: DFT over digit n_a (stride 4096), twiddle W_4096^{n_b*k_a}
//   stage 2: DFT over digit n_b (stride 64),   twiddle W_N^{n_c*(k_a+64*k_b)}
//   stage 3: DFT over digit n_c (stride 1),    scatter to natural order
// Forward: sgn=-1 (matches jnp.fft.fft). Inverse: sgn=+1, scale=1/N on stage 3.
// Block: 128 threads (4 waves), one 64x64 panel; wave w owns columns 16w..16w+15.
// ---------------------------------------------------------------------------
__global__ __launch_bounds__(128) void fft_stage_kernel(
    const float* __restrict__ srcR, const float* __restrict__ srcI,
    float* __restrict__ dstR, float* __restrict__ dstI,
    int stage, int inv, float scale)
{
  __shared__ float Xr [RADIX * RADIX];   // data panel, [j][l]
  __shared__ float Xi [RADIX * RADIX];
  __shared__ float DTr[RADIX * RADIX];   // DFT matrix transposed, [k][m]
  __shared__ float DTi[RADIX * RADIX];   // exactly 64 KB static LDS total

  const int tid = threadIdx.x;
  const int blk = blockIdx.x;
  const int sig = blk >> 6;                  // 64 panels per signal
  const int L0  = (blk & 63) * PANEL;
  const size_t sigOff = (size_t)sig * NFFT;
  const float sgn = inv ? 1.0f : -1.0f;

  // Build 64-point DFT matrix: D[m][k] = exp(sgn*2*pi*i*m*k/64), stored [k][m].
  for (int e = tid; e < RADIX * RADIX; e += 128) {
    const int k = e >> 6, m = e & 63;
    const float th = (2.0f * PI_F / 64.0f) * (float)((k * m) & 63);
    float sn, cs;
    __sincosf(th, &sn, &cs);
    DTr[e] = cs;
    DTi[e] = sgn * sn;
  }

  // ---- Stage the 64x64 panel into LDS ------------------------------------
  bool needElementLoad = true;
#if HAVE_TDM
  if (stage != 3) {
    // Stage 1: 2D tile, 64 rows (digit n_a) x 64 contiguous lines, pitch 4096.
    // Stage 2: one contiguous 4096-element row; both land in LDS as [j][l].
    const int waveId = __builtin_amdgcn_readfirstlane((int)(tid >> 5));
    if (waveId == 0) {
      const float* bR;
      const float* bI;
      uint32_t t0, t1, s0;
      if (stage == 1) {
        bR = srcR + sigOff + (size_t)L0;
        bI = srcI + sigOff + (size_t)L0;
        t0 = 64; t1 = 64; s0 = NLINES;
      } else {
        bR = srcR + sigOff + (size_t)L0 * 64;
        bI = srcI + sigOff + (size_t)L0 * 64;
        t0 = 4096; t1 = 1; s0 = NLINES;
      }
      tdm_load_2d(bR, lds_off(Xr), t0, t1, s0);
      tdm_load_2d(bI, lds_off(Xi), t0, t1, s0);
      __builtin_amdgcn_s_wait_tensorcnt(0);
    }
    needElementLoad = false;
  }
#endif
  if (needElementLoad) {
    for (int e = tid; e < RADIX * PANEL; e += 128) {
      int j, l;
      if (stage == 3) { l = e >> 6; j = e & 63; }   // stride-1 lines: j fast
      else            { j = e >> 6; l = e & 63; }   // line index fast (coalesced)
      const int Lg = L0 + l;
      size_t a;
      if (stage == 1)      a = sigOff + (size_t)j * NLINES + (size_t)Lg;
      else if (stage == 2) a = sigOff + (size_t)(Lg >> 6) * 4096 + (size_t)(Lg & 63) + (size_t)j * 64;
      else                 a = sigOff + (size_t)Lg * 64 + (size_t)j;
      const int x = j * RADIX + l;
#if USE_ASYNC
      {
        const unsigned long long gr = (unsigned long long)(uintptr_t)(srcR + a);
        const unsigned long long gi = (unsigned long long)(uintptr_t)(srcI + a);
        const uint32_t lr = lds_off(&Xr[x]);
        const uint32_t li = lds_off(&Xi[x]);
        asm volatile("global_load_async_to_lds_b32 %0, %1, off" :: "v"(lr), "v"(gr) : "memory");
        asm volatile("global_load_async_to_lds_b32 %0, %1, off" :: "v"(li), "v"(gi) : "memory");
      }
#else
      Xr[x] = srcR[a];
      Xi[x] = srcI[a];
#endif
    }
#if USE_ASYNC
    asm volatile("s_wait_asynccnt 0x0" ::: "memory");
#endif
  }
  __syncthreads();

  // ---- Complex GEMM: Y(64 x 16cols) = D(64x64) * X(64 x 16cols) per wave --
  const int lane = tid & 31;
  const int wave = tid >> 5;
  const int half = lane >> 4;                 // lane half selects K sub-pair
  const int colL = (wave << 4) + (lane & 15); // this lane's data column

  v8f accR[4], accI[4];
  const v8f vzero = {};
#pragma unroll
  for (int t = 0; t < 4; ++t) { accR[t] = vzero; accI[t] = vzero; }

#pragma unroll 2
  for (int kt = 0; kt < 16; ++kt) {           // K tiles of 4
    const int k0 = (kt << 2) + (half << 1);
    v2f br, bi;
    br.x = Xr[(k0    ) * RADIX + colL];
    br.y = Xr[(k0 + 1) * RADIX + colL];
    bi.x = Xi[(k0    ) * RADIX + colL];
    bi.y = Xi[(k0 + 1) * RADIX + colL];
#pragma unroll
    for (int mt = 0; mt < 4; ++mt) {          // M tiles of 16
      const int m = (mt << 4) + (lane & 15);
      v2f ar, ai, ain;
      ar.x = DTr[(k0    ) * RADIX + m];
      ar.y = DTr[(k0 + 1) * RADIX + m];
      ai.x = DTi[(k0    ) * RADIX + m];
      ai.y = DTi[(k0 + 1) * RADIX + m];
      ain.x = -ai.x; ain.y = -ai.y;
      // Yr += Dr*Xr - Di*Xi ; Yi += Dr*Xi + Di*Xr
      accR[mt] = __builtin_amdgcn_wmma_f32_16x16x4_f32(false, ar,  false, br, (short)0, accR[mt], false, false);
      accR[mt] = __builtin_amdgcn_wmma_f32_16x16x4_f32(false, ain, false, bi, (short)0, accR[mt], false, false);
      accI[mt] = __builtin_amdgcn_wmma_f32_16x16x4_f32(false, ar,  false, bi, (short)0, accI[mt], false, false);
      accI[mt] = __builtin_amdgcn_wmma_f32_16x16x4_f32(false, ai,  false, br, (short)0, accI[mt], false, false);
    }
  }

  // ---- Twiddle + store. C/D: vgpr v, lane<16 -> M=mt*16+v, lane>=16 -> +8 -
  const int Lg = L0 + colL;
  const int la = Lg >> 6;     // slow digit of the line index
  const int lb = Lg & 63;     // fast digit of the line index
#pragma unroll
  for (int mt = 0; mt < 4; ++mt) {
#pragma unroll
    for (int v = 0; v < 8; ++v) {
      const int m = (mt << 4) + (half << 3) + v;
      const float yr = accR[mt][v];
      const float yi = accI[mt][v];
      float cs = 1.0f, sn = 0.0f;
      if (stage == 1) {
        const float ang = sgn * (2.0f * PI_F / 4096.0f) * (float)(la * m);
        __sincosf(ang, &sn, &cs);
      } else if (stage == 2) {
        const float ang = sgn * (2.0f * PI_F / 262144.0f) * (float)(lb * (la + (m << 6)));
        __sincosf(ang, &sn, &cs);
      }
      const float outR = (yr * cs - yi * sn) * scale;
      const float outI = (yr * sn + yi * cs) * scale;
      size_t a;
      if (stage == 1)      a = sigOff + (size_t)m * 4096 + (size_t)Lg;
      else if (stage == 2) a = sigOff + (size_t)la * 4096 + (size_t)lb + (size_t)m * 64;
      else                 a = sigOff + (size_t)la + (size_t)lb * 64 + (size_t)m * 4096; // natural order
      dstR[a] = outR;
      dstI[a] = outI;
    }
  }
}

// ---- pointwise kernels -----------------------------------------------------

__global__ __launch_bounds__(256) void pack_kernel(
    const float* __restrict__ u_re, const float* __restrict__ u_im,
    float* __restrict__ wR, float* __restrict__ wI)
{
  const size_t idx = (size_t)blockIdx.x * 256 + threadIdx.x;   // NSIG*NFFT
  const int sig = (int)(idx >> 18);
  const int n   = (int)(idx & (NFFT - 1));
  const int b = sig >> 1, p = sig & 1;
  const size_t src = ((size_t)b * NFFT + n) * 2 + p;
  wR[idx] = u_re[src];
  wI[idx] = u_im[src];
}

__global__ __launch_bounds__(256) void hmul_kernel(
    float* __restrict__ yR, float* __restrict__ yI,
    const float* __restrict__ H_re, const float* __restrict__ H_im, int s)
{
  const size_t idx = (size_t)blockIdx.x * 256 + threadIdx.x;   // NSIG*NFFT
  const int n = (int)(idx & (NFFT - 1));
  const float hr = H_re[(size_t)s * NFFT + n];
  const float hi = H_im[(size_t)s * NFFT + n];
  const float yr = yR[idx], yi = yI[idx];
  yR[idx] = yr * hr - yi * hi;
  yI[idx] = yr * hi + yi * hr;
}

__global__ __launch_bounds__(256) void nonlinear_kernel(
    float* __restrict__ uR, float* __restrict__ uI,
    const float* __restrict__ scales, const float* __restrict__ phi, int s)
{
  const size_t idx = (size_t)blockIdx.x * 256 + threadIdx.x;   // BATCH*NFFT
  const int b = (int)(idx >> 18);
  const int n = (int)(idx & (NFFT - 1));
  const size_t i0 = (size_t)(b * 2) * NFFT + n;
  const size_t i1 = i0 + NFFT;
  const float r0 = uR[i0], q0 = uI[i0];
  const float r1 = uR[i1], q1 = uI[i1];
  const float pw = r0 * r0 + q0 * q0 + r1 * r1 + q1 * q1;
  const float ang = -(scales[s] * phi[0]) * pw;
  float sn, cs;
  __sincosf(ang, &sn, &cs);
  uR[i0] = r0 * cs - q0 * sn;  uI[i0] = r0 * sn + q0 * cs;
  uR[i1] = r1 * cs - q1 * sn;  uI[i1] = r1 * sn + q1 * cs;
}

__global__ __launch_bounds__(256) void conv_kernel(
    const float* __restrict__ uR, const float* __restrict__ uI,
    const float* __restrict__ wr, const float* __restrict__ br,
    const float* __restrict__ wi, const float* __restrict__ bi,
    float* __restrict__ out)
{
  const int idx = blockIdx.x * 256 + threadIdx.x;   // BATCH * 2 * LOUT
  const int t = idx & (LOUT - 1);
  const int o = (idx >> 15) & 1;
  const int b = idx >> 16;
  float accRe = br[o] - bi[o];
  float accIm = br[o] + bi[o];
#pragma unroll
  for (int i = 0; i < 2; ++i) {
    const float* xr = uR + (size_t)(b * 2 + i) * NFFT + (size_t)t * 8;
    const float* xi = uI + (size_t)(b * 2 + i) * NFFT + (size_t)t * 8;
#pragma unroll
    for (int k = 0; k < 8; ++k) {
      const float wrv = wr[(o * 2 + i) * 8 + k];
      const float wiv = wi[(o * 2 + i) * 8 + k];
      const float r = xr[k], q = xi[k];
      accRe += wrv * r - wiv * q;
      accIm += wrv * q + wiv * r;
    }
  }
  const size_t oidx = (((size_t)b * LOUT + t) * 2 + o) * 2;
  out[oidx]     = accRe;
  out[oidx + 1] = accIm;
}

// ---- launcher --------------------------------------------------------------

extern "C" void kernel_launch(void* const* d_in, const int* in_sizes, int n_in,
                              void* d_out, int out_size, void* d_ws, size_t ws_size,
                              hipStream_t stream) {
  const float* u_re   = (const float*)d_in[0];
  const float* u_im   = (const float*)d_in[1];
  const float* H_re   = (const float*)d_in[2];
  const float* H_im   = (const float*)d_in[3];
  const float* scales = (const float*)d_in[4];
  const float* phi    = (const float*)d_in[5];
  const float* wr     = (const float*)d_in[6];
  const float* br     = (const float*)d_in[7];
  const float* wi     = (const float*)d_in[8];
  const float* bi     = (const float*)d_in[9];
  float* out = (float*)d_out;

  // Scratch: two ping-pong planar complex buffers, 4 x 16.78 MB = 67.1 MB.
  const size_t planeElems = (size_t)NSIG * NFFT;
  float* bufAR = (float*)d_ws;
  float* bufAI = bufAR + planeElems;
  float* bufBR = bufAI + planeElems;
  float* bufBI = bufBR + planeElems;

  const int packBlocks  = (int)(planeElems / 256);          // 16384
  const int stageBlocks = NSIG * (NLINES / PANEL);          // 1024
  const int nlBlocks    = (int)((size_t)BATCH * NFFT / 256);// 8192
  const int convBlocks  = (BATCH * 2 * LOUT) / 256;         // 2048
  const float invN = 1.0f / (float)NFFT;

  pack_kernel<<<packBlocks, 256, 0, stream>>>(u_re, u_im, bufAR, bufAI);

  for (int s = 0; s < NSTEP; ++s) {
    // forward FFT (natural-order output via stage-3 scatter)
    fft_stage_kernel<<<stageBlocks, 128, 0, stream>>>(bufAR, bufAI, bufBR, bufBI, 1, 0, 1.0f);
    fft_stage_kernel<<<stageBlocks, 128, 0, stream>>>(bufBR, bufBI, bufAR, bufAI, 2, 0, 1.0f);
    fft_stage_kernel<<<stageBlocks, 128, 0, stream>>>(bufAR, bufAI, bufBR, bufBI, 3, 0, 1.0f);
    // frequency-domain filter
    hmul_kernel<<<packBlocks, 256, 0, stream>>>(bufBR, bufBI, H_re, H_im, s);
    // inverse FFT (1/N folded into last stage)
    fft_stage_kernel<<<stageBlocks, 128, 0, stream>>>(bufBR, bufBI, bufAR, bufAI, 1, 1, 1.0f);
    fft_stage_kernel<<<stageBlocks, 128, 0, stream>>>(bufAR, bufAI, bufBR, bufBI, 2, 1, 1.0f);
    fft_stage_kernel<<<stageBlocks, 128, 0, stream>>>(bufBR, bufBI, bufAR, bufAI, 3, 1, invN);
    // nonlinear phase rotation (couples the two pols)
    nonlinear_kernel<<<nlBlocks, 256, 0, stream>>>(bufAR, bufAI, scales, phi, s);
  }

  conv_kernel<<<convBlocks, 256, 0, stream>>>(bufAR, bufAI, wr, br, wi, bi, out);
}